// Hopfield_Kuramoto_network_59682865545242
// MI455X (gfx1250) — compile-verified
//
#include <hip/hip_runtime.h>
#include <math.h>

// Problem sizes (fixed by the reference)
#define N_NODES 8192
#define D_OSC   64
#define E_EDGES 1048576

typedef __attribute__((ext_vector_type(2))) float v2f;
typedef __attribute__((ext_vector_type(8))) float v8f;

__device__ __forceinline__ float wave_reduce_sum(float v) {
  // wave32 butterfly; every lane ends with the full sum
  #pragma unroll
  for (int m = 16; m >= 1; m >>= 1) v += __shfl_xor(v, m, 32);
  return v;
}

// ---------------------------------------------------------------------------
// 0) zero the workspace accumulators (re-run every launch: harness poisons ws)
// ---------------------------------------------------------------------------
__global__ __launch_bounds__(256) void k_zero(float* __restrict__ p, int n) {
  int i = blockIdx.x * 256 + threadIdx.x;
  if (i < n) p[i] = 0.0f;
}

// ---------------------------------------------------------------------------
// 1) normalize state_K rows -> skn ; g = tanh(state_H)
//    one wave per row (lane l holds elements l and l+32)
// ---------------------------------------------------------------------------
__global__ __launch_bounds__(256) void k_norm_tanh(const float* __restrict__ sk,
                                                   const float* __restrict__ sh,
                                                   float* __restrict__ skn,
                                                   float* __restrict__ gv) {
  int row  = blockIdx.x * 8 + (threadIdx.x >> 5);
  int lane = threadIdx.x & 31;
  const float* r = sk + (size_t)row * D_OSC;
  float a = r[lane];
  float b = r[lane + 32];
  float ss  = wave_reduce_sum(a * a + b * b);
  float inv = rsqrtf(ss);
  float* o = skn + (size_t)row * D_OSC;
  o[lane]      = a * inv;
  o[lane + 32] = b * inv;

  int gi = blockIdx.x * 256 + threadIdx.x;   // first 32 blocks cover 8192
  if (gi < N_NODES) gv[gi] = tanhf(sh[gi]);
}

// ---------------------------------------------------------------------------
// 2) Kuramoto edges: s = <skn[j0], skn[j1]>, scatter w_K*s*skn into fK accum
//    one wave per edge; all operands are L2-resident (2 MB)
// ---------------------------------------------------------------------------
__global__ __launch_bounds__(256) void k_edge_K(const int* __restrict__ ind,
                                                const float* __restrict__ wK,
                                                const float* __restrict__ skn,
                                                float* __restrict__ fK) {
  int e    = blockIdx.x * 8 + (threadIdx.x >> 5);
  int lane = threadIdx.x & 31;
  int j0 = ind[2 * e];
  int j1 = ind[2 * e + 1];
  const float* r0 = skn + (size_t)j0 * D_OSC;
  const float* r1 = skn + (size_t)j1 * D_OSC;
  float a0 = r0[lane], a1 = r0[lane + 32];
  float b0 = r1[lane], b1 = r1[lane + 32];
  float s = wave_reduce_sum(a0 * b0 + a1 * b1);
  float c = wK[e] * s;
  atomicAdd(&fK[(size_t)j0 * D_OSC + lane],       c * b0);
  atomicAdd(&fK[(size_t)j0 * D_OSC + lane + 32],  c * b1);
  atomicAdd(&fK[(size_t)j1 * D_OSC + lane],       c * a0);
  atomicAdd(&fK[(size_t)j1 * D_OSC + lane + 32],  c * a1);
}

// ---------------------------------------------------------------------------
// 3) Hopfield-Kuramoto coupling edges
// ---------------------------------------------------------------------------
__global__ __launch_bounds__(256) void k_edge_HK(const int* __restrict__ ind,
                                                 const float* __restrict__ whk,
                                                 const float* __restrict__ skn,
                                                 const float* __restrict__ gv,
                                                 const float* __restrict__ kH,
                                                 const float* __restrict__ kK,
                                                 float* __restrict__ fH,
                                                 float* __restrict__ fK) {
  int e    = blockIdx.x * 8 + (threadIdx.x >> 5);
  int lane = threadIdx.x & 31;
  int i0 = ind[2 * e];
  int i1 = ind[2 * e + 1];
  const float* r0 = skn + (size_t)i0 * D_OSC;
  const float* r1 = skn + (size_t)i1 * D_OSC;
  float a0 = r0[lane], a1 = r0[lane + 32];
  float b0 = r1[lane], b1 = r1[lane + 32];
  float Gram = wave_reduce_sum(a0 * b0 + a1 * b1);
  float w  = whk[e];
  float g0 = gv[i0];
  float g1 = gv[i1];
  float invKH = 1.0f / kH[0];
  float invKK = 1.0f / kK[0];
  if (lane == 0) {
    atomicAdd(&fH[i0], Gram * w * g1 * invKH);
    atomicAdd(&fH[i1], Gram * w * g0 * invKH);
  }
  float Gc = -(g0 * g1) * w * invKK;
  atomicAdd(&fK[(size_t)i0 * D_OSC + lane],      Gc * b0);
  atomicAdd(&fK[(size_t)i0 * D_OSC + lane + 32], Gc * b1);
  atomicAdd(&fK[(size_t)i1 * D_OSC + lane],      Gc * a0);
  atomicAdd(&fK[(size_t)i1 * D_OSC + lane + 32], Gc * a1);
}

// ---------------------------------------------------------------------------
// 4) FUSED y = W@g and z = W^T@g in ONE pass over W (256 MB read once).
//
//    Workgroup (8 waves) owns a 128-row x 512-col region. Wave w owns a
//    16-row strip and walks 32 tiles of 16x16:
//      * y: 4x V_WMMA_F32_16X16X4_F32 into a persistent accumulator
//           (B = g[c..c+16] broadcast across all 16 columns).
//      * tile is spilled to a PRIVATE per-wave LDS slab (row stride 18
//        dwords -> conflict-free stores and transposed loads; same-wave DS
//        ops are in-order so no barrier is needed).
//      * z: transposed reads from LDS feed 4 more wmmas with
//           B = g[r0..r0+16] (constant per wave); the 16 per-tile z partials
//           are ds_add_f32'ed into a workgroup staging buffer.
//    Global flush: 16 y-atomics per wave + 512 z-atomics per workgroup.
// ---------------------------------------------------------------------------
__global__ __launch_bounds__(256) void k_gemv_fused(const float* __restrict__ W,
                                                    const float* __restrict__ gv,
                                                    float* __restrict__ y,
                                                    float* __restrict__ z) {
  __shared__ float slab[8 * 288];     // 16 rows x 18-dword stride per wave
  __shared__ float zstage[512];

  const int wv   = threadIdx.x >> 5;  // wave in workgroup 0..7
  const int lane = threadIdx.x & 31;
  const int hl   = lane & 15;         // half-lane id
  const int koff = (lane < 16) ? 0 : 2;

  const int C0 = blockIdx.x * 512;    // column chunk (16 chunks)
  const int r0 = blockIdx.y * 128 + wv * 16;   // wave's row base

  // zero the z staging buffer
  zstage[threadIdx.x]       = 0.0f;
  zstage[threadIdx.x + 256] = 0.0f;
  __syncthreads();

  float* sl = slab + wv * 288;

  // B operands for the z-path wmmas are fixed per wave (rows r0..r0+15)
  v2f bz0 = *(const v2f*)(gv + r0 +  0 + koff);
  v2f bz1 = *(const v2f*)(gv + r0 +  4 + koff);
  v2f bz2 = *(const v2f*)(gv + r0 +  8 + koff);
  v2f bz3 = *(const v2f*)(gv + r0 + 12 + koff);

  const float* wrow = W + (size_t)(r0 + hl) * N_NODES + koff;

  v8f yacc = {};
  for (int t = 0; t < 32; ++t) {
    const int c = C0 + t * 16;
    const float* wp = wrow + c;
    // tile registers, A-layout for the y-path (16 rows x 4-col chunks)
    v2f a0 = *(const v2f*)(wp +  0);
    v2f a1 = *(const v2f*)(wp +  4);
    v2f a2 = *(const v2f*)(wp +  8);
    v2f a3 = *(const v2f*)(wp + 12);
    v2f by0 = *(const v2f*)(gv + c +  0 + koff);
    v2f by1 = *(const v2f*)(gv + c +  4 + koff);
    v2f by2 = *(const v2f*)(gv + c +  8 + koff);
    v2f by3 = *(const v2f*)(gv + c + 12 + koff);

    yacc = __builtin_amdgcn_wmma_f32_16x16x4_f32(false, a0, false, by0, (short)0, yacc, false, false);
    yacc = __builtin_amdgcn_wmma_f32_16x16x4_f32(false, a1, false, by1, (short)0, yacc, false, false);
    yacc = __builtin_amdgcn_wmma_f32_16x16x4_f32(false, a2, false, by2, (short)0, yacc, false, false);
    yacc = __builtin_amdgcn_wmma_f32_16x16x4_f32(false, a3, false, by3, (short)0, yacc, false, false);

    // spill tile row-major into the wave-private slab
    float* sr = sl + hl * 18 + koff;
    *(v2f*)(sr +  0) = a0;
    *(v2f*)(sr +  4) = a1;
    *(v2f*)(sr +  8) = a2;
    *(v2f*)(sr + 12) = a3;

    // transposed reads: A'[m,k] = T[4j+k, m]
    v2f az0, az1, az2, az3;
    az0[0] = sl[( 0 + koff) * 18 + hl];  az0[1] = sl[( 1 + koff) * 18 + hl];
    az1[0] = sl[( 4 + koff) * 18 + hl];  az1[1] = sl[( 5 + koff) * 18 + hl];
    az2[0] = sl[( 8 + koff) * 18 + hl];  az2[1] = sl[( 9 + koff) * 18 + hl];
    az3[0] = sl[(12 + koff) * 18 + hl];  az3[1] = sl[(13 + koff) * 18 + hl];

    v8f zt = {};
    zt = __builtin_amdgcn_wmma_f32_16x16x4_f32(false, az0, false, bz0, (short)0, zt, false, false);
    zt = __builtin_amdgcn_wmma_f32_16x16x4_f32(false, az1, false, bz1, (short)0, zt, false, false);
    zt = __builtin_amdgcn_wmma_f32_16x16x4_f32(false, az2, false, bz2, (short)0, zt, false, false);
    zt = __builtin_amdgcn_wmma_f32_16x16x4_f32(false, az3, false, bz3, (short)0, zt, false, false);

    // stage the 16 z partials for cols c..c+15 (lane0 -> 0-7, lane16 -> 8-15)
    if (hl == 0) {
      int zb = t * 16 + ((lane == 16) ? 8 : 0);
      #pragma unroll
      for (int v = 0; v < 8; ++v) atomicAdd(&zstage[zb + v], zt[v]);
    }
  }

  // flush y partials (persistent accumulator, rows fixed per wave)
  if (hl == 0) {
    int base = r0 + ((lane == 16) ? 8 : 0);
    #pragma unroll
    for (int v = 0; v < 8; ++v) atomicAdd(&y[base + v], yacc[v]);
  }

  __syncthreads();
  // flush z staging buffer once per workgroup
  atomicAdd(&z[C0 + threadIdx.x],       zstage[threadIdx.x]);
  atomicAdd(&z[C0 + threadIdx.x + 256], zstage[threadIdx.x + 256]);
}

// ---------------------------------------------------------------------------
// 5) finalize f_H = edge_acc + (y+z)/2 - state_H + bias_H
// ---------------------------------------------------------------------------
__global__ __launch_bounds__(256) void k_fin_H(const float* __restrict__ fHacc,
                                               const float* __restrict__ y,
                                               const float* __restrict__ z,
                                               const float* __restrict__ sh,
                                               const float* __restrict__ bias,
                                               float* __restrict__ out) {
  int i = blockIdx.x * 256 + threadIdx.x;
  if (i < N_NODES)
    out[i] = fHacc[i] + 0.5f * (y[i] + z[i]) - sh[i] + bias[i];
}

// ---------------------------------------------------------------------------
// 6) finalize f_K = -raw + skn * <skn, raw>   (one wave per row)
// ---------------------------------------------------------------------------
__global__ __launch_bounds__(256) void k_fin_K(const float* __restrict__ fKacc,
                                               const float* __restrict__ skn,
                                               float* __restrict__ out) {
  int row  = blockIdx.x * 8 + (threadIdx.x >> 5);
  int lane = threadIdx.x & 31;
  size_t base = (size_t)row * D_OSC;
  float r0 = fKacc[base + lane], r1 = fKacc[base + lane + 32];
  float s0 = skn[base + lane],   s1 = skn[base + lane + 32];
  float dot = wave_reduce_sum(s0 * r0 + s1 * r1);
  out[base + lane]      = -r0 + s0 * dot;
  out[base + lane + 32] = -r1 + s1 * dot;
}

// ---------------------------------------------------------------------------
extern "C" void kernel_launch(void* const* d_in, const int* in_sizes, int n_in,
                              void* d_out, int out_size, void* d_ws, size_t ws_size,
                              hipStream_t stream) {
  const float* state_H    = (const float*)d_in[0];
  const float* state_K    = (const float*)d_in[1];
  const int*   ind_K      = (const int*)d_in[2];
  const int*   ind_HK     = (const int*)d_in[3];
  const float* kappa_K    = (const float*)d_in[4];
  const float* kappa_H    = (const float*)d_in[5];
  const float* weights_H  = (const float*)d_in[6];
  const float* bias_H     = (const float*)d_in[7];
  const float* weights_HK = (const float*)d_in[8];
  const float* w_K        = (const float*)d_in[9];

  float* out_fH = (float*)d_out;                 // 8192
  float* out_fK = out_fH + N_NODES;              // 8192*64

  // workspace layout (floats): [fK_acc | fH_acc | y | z | skn | g]
  float* ws     = (float*)d_ws;
  float* fK_acc = ws;                                   // 524288
  float* fH_acc = fK_acc + (size_t)N_NODES * D_OSC;     // 8192
  float* yv     = fH_acc + N_NODES;                     // 8192
  float* zv     = yv + N_NODES;                         // 8192
  float* skn    = zv + N_NODES;                         // 524288
  float* gv     = skn + (size_t)N_NODES * D_OSC;        // 8192

  const int nAcc = N_NODES * D_OSC + 3 * N_NODES;       // 548864 (contiguous)

  k_zero<<<(nAcc + 255) / 256, 256, 0, stream>>>(ws, nAcc);

  k_norm_tanh<<<N_NODES / 8, 256, 0, stream>>>(state_K, state_H, skn, gv);

  k_edge_K<<<E_EDGES / 8, 256, 0, stream>>>(ind_K, w_K, skn, fK_acc);

  k_edge_HK<<<E_EDGES / 8, 256, 0, stream>>>(ind_HK, weights_HK, skn, gv,
                                             kappa_H, kappa_K, fH_acc, fK_acc);

  // single pass over W: grid = (col chunks of 512) x (row chunks of 128)
  k_gemv_fused<<<dim3(16, 64), 256, 0, stream>>>(weights_H, gv, yv, zv);

  k_fin_H<<<(N_NODES + 255) / 256, 256, 0, stream>>>(fH_acc, yv, zv,
                                                     state_H, bias_H, out_fH);
  k_fin_K<<<N_NODES / 8, 256, 0, stream>>>(fK_acc, skn, out_fK);
}